// GCN_66443144069640
// MI455X (gfx1250) — compile-verified
//
#include <hip/hip_runtime.h>
#include <math.h>

#define EMB 12
#define NCLS 17

typedef __attribute__((ext_vector_type(2))) float v2f;
typedef __attribute__((ext_vector_type(8))) float v8f;

// ---- helpers -------------------------------------------------------------

__device__ __forceinline__ void fatomicAdd(float* p, float v) {
  // native global_atomic_add_f32 (no CAS loop)
  unsafeAtomicAdd(p, v);
}

// monotonic float->uint key so unsigned atomicMax == float max
__device__ __forceinline__ unsigned fkey(float v) {
  unsigned u = __float_as_uint(v);
  return (u & 0x80000000u) ? ~u : (u | 0x80000000u);
}
__device__ __forceinline__ float funkey(unsigned k) {
  unsigned u = (k & 0x80000000u) ? (k & 0x7FFFFFFFu) : ~k;
  return __uint_as_float(u);
}

// ---- degree / norm precompute -------------------------------------------

__global__ void k_init_deg(float* deg, int n) {
  int i = blockIdx.x * blockDim.x + threadIdx.x;
  if (i < n) deg[i] = 1.0f;  // self-loop
}

__global__ void k_deg(const long long* __restrict__ ei, float* __restrict__ deg, int E) {
  int e = blockIdx.x * blockDim.x + threadIdx.x;
  if (e >= E) return;
  int c = (int)ei[(long long)E + e];  // col
  fatomicAdd(&deg[c], 1.0f);
}

__global__ void k_dinv(float* __restrict__ dinv, float* __restrict__ dinv2, int n) {
  int i = blockIdx.x * blockDim.x + threadIdx.x;
  if (i >= n) return;
  float d = 1.0f / sqrtf(dinv[i]);
  dinv[i] = d;
  dinv2[i] = d * d;
}

__global__ void k_norm(const long long* __restrict__ ei, const float* __restrict__ dinv,
                       float* __restrict__ norm, int E) {
  int e = blockIdx.x * blockDim.x + threadIdx.x;
  if (e >= E) return;
  int r = (int)ei[e];
  int c = (int)ei[(long long)E + e];
  norm[e] = dinv[r] * dinv[c];
}

// ---- dense transform: t = (tanh?)(in) @ W + b via V_WMMA_F32_16X16X4_F32 --
// One wave per 16-node tile. A: 16x4 f32 (lane halves hold K pairs),
// B: 4x16 f32 (W columns striped over lanes), C/D: 16x16 f32.
// Columns of D are independent, so B/bias columns 12..15 may hold clamped
// garbage (never read) -> all loads unconditional, no EXEC churn.

template <int FIN, bool DO_TANH>
__global__ void k_dense(const float* __restrict__ in, const float* __restrict__ W,
                        const float* __restrict__ bias, float* __restrict__ t,
                        int n, int ntiles) {
  int lane = threadIdx.x & 31;
  int wave = threadIdx.x >> 5;
  int tile = blockIdx.x * (blockDim.x >> 5) + wave;
  if (tile >= ntiles) return;

  int base = tile * 16;
  int m    = lane & 15;   // row (A) / column (B,C,D) index
  int half = lane >> 4;   // 0: K={0,1}, 1: K={2,3}
  int mc   = (m < EMB) ? m : (EMB - 1);  // clamped column for B/bias loads

  int node = base + m;
  if (node > n - 1) node = n - 1;  // clamp: loads stay valid & unpredicated

  float bv = bias[mc];
  v8f c;
#pragma unroll
  for (int r = 0; r < 8; ++r) c[r] = bv;

#pragma unroll
  for (int kb = 0; kb < FIN; kb += 4) {
    int ka = kb + half * 2;  // even -> 8B-aligned pair load
    v2f a = *(const v2f*)(in + (long long)node * FIN + ka);
    if (DO_TANH) { a[0] = tanhf(a[0]); a[1] = tanhf(a[1]); }
    v2f b;
    b[0] = W[(ka + 0) * EMB + mc];
    b[1] = W[(ka + 1) * EMB + mc];
    c = __builtin_amdgcn_wmma_f32_16x16x4_f32(false, a, false, b, (short)0, c,
                                              false, false);
  }

  // D layout: VGPR r holds row M = half*8 + r, column = m.
  // Store base precomputed -> 8 stores with constant immediate offsets.
  float* tp = t + (long long)(base + half * 8) * EMB + m;
  if (base + 16 <= n) {        // wave-uniform fast path (always true for n%16==0)
    if (m < EMB) {
#pragma unroll
      for (int r = 0; r < 8; ++r) tp[r * EMB] = c[r];
    }
  } else {                     // tail tile (generic safety)
    if (m < EMB) {
#pragma unroll
      for (int r = 0; r < 8; ++r)
        if (base + half * 8 + r < n) tp[r * EMB] = c[r];
    }
  }
}

// ---- self-loop contribution: agg[i] = dinv2[i] * t[i] --------------------

__global__ void k_selfloop(const float* __restrict__ t, const float* __restrict__ dinv2,
                           float* __restrict__ agg, int total) {
  int i = blockIdx.x * blockDim.x + threadIdx.x;
  if (i >= total) return;
  agg[i] = dinv2[i / EMB] * t[i];
}

// ---- edge scatter: agg[col] += norm * t[row] (L2-resident FP atomics) ----

__global__ void k_scatter(const long long* __restrict__ ei, const float* __restrict__ norm,
                          const float* __restrict__ t, float* __restrict__ agg, int E) {
  int e = blockIdx.x * blockDim.x + threadIdx.x;
  if (e >= E) return;
  int r = (int)ei[e];
  int c = (int)ei[(long long)E + e];
  float nv = norm[e];
  const float4* tr = (const float4*)(t + (long long)r * EMB);  // 48B rows, 16B aligned
  float* ag = agg + (long long)c * EMB;
  float4 v0 = tr[0], v1 = tr[1], v2 = tr[2];
  fatomicAdd(ag + 0, nv * v0.x);  fatomicAdd(ag + 1, nv * v0.y);
  fatomicAdd(ag + 2, nv * v0.z);  fatomicAdd(ag + 3, nv * v0.w);
  fatomicAdd(ag + 4, nv * v1.x);  fatomicAdd(ag + 5, nv * v1.y);
  fatomicAdd(ag + 6, nv * v1.z);  fatomicAdd(ag + 7, nv * v1.w);
  fatomicAdd(ag + 8, nv * v2.x);  fatomicAdd(ag + 9, nv * v2.y);
  fatomicAdd(ag + 10, nv * v2.z); fatomicAdd(ag + 11, nv * v2.w);
}

// ---- pooling -------------------------------------------------------------

__global__ void k_initpool(unsigned* gmaxu, float* gsum, float* cnt, int G) {
  int i = blockIdx.x * blockDim.x + threadIdx.x;
  if (i < G * EMB) { gmaxu[i] = 0u; gsum[i] = 0.0f; }
  if (i < G) cnt[i] = 0.0f;
}

__global__ void k_pool(const float* __restrict__ agg, const long long* __restrict__ batch,
                       unsigned* __restrict__ gmaxu, float* __restrict__ gsum,
                       float* __restrict__ cnt, int n) {
  int i = blockIdx.x * blockDim.x + threadIdx.x;
  if (i >= n) return;
  int g = (int)batch[i];
  fatomicAdd(&cnt[g], 1.0f);
  const float* a = agg + (long long)i * EMB;
#pragma unroll
  for (int f = 0; f < EMB; ++f) {
    float v = tanhf(a[f]);
    fatomicAdd(&gsum[g * EMB + f], v);
    atomicMax(&gmaxu[g * EMB + f], fkey(v));
  }
}

// ---- head: hidden = [gmax, gmean]; out = hidden @ Wout + bout ------------

__global__ void k_head(const unsigned* __restrict__ gmaxu, const float* __restrict__ gsum,
                       const float* __restrict__ cnt, const float* __restrict__ Wout,
                       const float* __restrict__ bout, float* __restrict__ out, int G) {
  int g = blockIdx.x * blockDim.x + threadIdx.x;
  if (g >= G) return;
  float hidden[2 * EMB];
  float c = fmaxf(cnt[g], 1.0f);
#pragma unroll
  for (int f = 0; f < EMB; ++f) {
    hidden[f]       = funkey(gmaxu[g * EMB + f]);
    hidden[EMB + f] = gsum[g * EMB + f] / c;
  }
#pragma unroll
  for (int j = 0; j < NCLS; ++j) {
    float acc = bout[j];
#pragma unroll
    for (int k = 0; k < 2 * EMB; ++k) acc += hidden[k] * Wout[k * NCLS + j];
    out[g * NCLS + j] = acc;
  }
  float* hid = out + (long long)G * NCLS;
#pragma unroll
  for (int k = 0; k < 2 * EMB; ++k) hid[g * 2 * EMB + k] = hidden[k];
}

// ---- launcher ------------------------------------------------------------

extern "C" void kernel_launch(void* const* d_in, const int* in_sizes, int n_in,
                              void* d_out, int out_size, void* d_ws, size_t ws_size,
                              hipStream_t stream) {
  const float*     x     = (const float*)d_in[0];
  const long long* ei    = (const long long*)d_in[1];
  const long long* batch = (const long long*)d_in[2];
  const float* W0 = (const float*)d_in[3];  const float* b0 = (const float*)d_in[4];
  const float* W1 = (const float*)d_in[5];  const float* b1 = (const float*)d_in[6];
  const float* W2 = (const float*)d_in[7];  const float* b2 = (const float*)d_in[8];
  const float* W3 = (const float*)d_in[9];  const float* b3 = (const float*)d_in[10];
  const float* Wout = (const float*)d_in[11];
  const float* bout = (const float*)d_in[12];
  float* out = (float*)d_out;

  const int n = in_sizes[0] / 4;       // 400000
  const int E = in_sizes[1] / 2;       // 6400000
  const int G = out_size / (NCLS + 2 * EMB);  // 1000

  // workspace layout (256B aligned slabs)
  char* ws = (char*)d_ws;
  size_t off = 0;
  auto take = [&](size_t bytes) -> void* {
    void* p = ws + off;
    off += (bytes + 255) & ~(size_t)255;
    return p;
  };
  float*    norm  = (float*)take((size_t)E * 4);
  float*    dinv  = (float*)take((size_t)n * 4);   // deg, then deg^-0.5 in place
  float*    dinv2 = (float*)take((size_t)n * 4);
  float*    t     = (float*)take((size_t)n * EMB * 4);
  float*    agg   = (float*)take((size_t)n * EMB * 4);
  unsigned* gmaxu = (unsigned*)take((size_t)G * EMB * 4);
  float*    gsum  = (float*)take((size_t)G * EMB * 4);
  float*    cnt   = (float*)take((size_t)G * 4);
  (void)ws_size; (void)n_in;

  const int B = 256;
  const int nB  = (n + B - 1) / B;
  const int eB  = (E + B - 1) / B;
  const int neB = (n * EMB + B - 1) / B;
  const int ntiles = (n + 15) / 16;
  const int dB = (ntiles + 7) / 8;  // 8 waves/block

  // degree + norm (once)
  k_init_deg<<<nB, B, 0, stream>>>(dinv, n);
  k_deg<<<eB, B, 0, stream>>>(ei, dinv, E);
  k_dinv<<<nB, B, 0, stream>>>(dinv, dinv2, n);
  k_norm<<<eB, B, 0, stream>>>(ei, dinv, norm, E);

  // layer 0 (Fin=4, raw x), layers 1..3 (Fin=12, tanh on input)
  k_dense<4, false><<<dB, B, 0, stream>>>(x, W0, b0, t, n, ntiles);
  k_selfloop<<<neB, B, 0, stream>>>(t, dinv2, agg, n * EMB);
  k_scatter<<<eB, B, 0, stream>>>(ei, norm, t, agg, E);

  k_dense<EMB, true><<<dB, B, 0, stream>>>(agg, W1, b1, t, n, ntiles);
  k_selfloop<<<neB, B, 0, stream>>>(t, dinv2, agg, n * EMB);
  k_scatter<<<eB, B, 0, stream>>>(ei, norm, t, agg, E);

  k_dense<EMB, true><<<dB, B, 0, stream>>>(agg, W2, b2, t, n, ntiles);
  k_selfloop<<<neB, B, 0, stream>>>(t, dinv2, agg, n * EMB);
  k_scatter<<<eB, B, 0, stream>>>(ei, norm, t, agg, E);

  k_dense<EMB, true><<<dB, B, 0, stream>>>(agg, W3, b3, t, n, ntiles);
  k_selfloop<<<neB, B, 0, stream>>>(t, dinv2, agg, n * EMB);
  k_scatter<<<eB, B, 0, stream>>>(ei, norm, t, agg, E);

  // pooling (final tanh fused) + head
  k_initpool<<<(G * EMB + B - 1) / B, B, 0, stream>>>(gmaxu, gsum, cnt, G);
  k_pool<<<nB, B, 0, stream>>>(agg, batch, gmaxu, gsum, cnt, n);
  k_head<<<(G + 63) / 64, 64, 0, stream>>>(gmaxu, gsum, cnt, Wout, bout, out, G);
}